// LNon_369367188002
// MI455X (gfx1250) — compile-verified
//
#include <hip/hip_runtime.h>
#include <hip/hip_bf16.h>
#include <math.h>

// ---------------------------------------------------------------------------
// LNon: global-norm -> table-driven pointwise transform -> global-norm.
// 3 streaming passes over 128MB + tiny table/finalize kernels.
// Deterministic fixed-order reductions (no float atomics).
// WMMA f32 16x16x4 used (correctly, per ISA VGPR layouts) to build the
// 11-entry effective tables  eff[i] = sum_j param[i]*softmax(w)[i,j].
// Hot loop uses branch-free hardware-trans math (v_exp/v_rcp/v_sin/v_cos)
// instead of libm tanhf / IEEE division.
// ---------------------------------------------------------------------------

typedef __attribute__((ext_vector_type(2))) float v2f;
typedef __attribute__((ext_vector_type(4))) float v4f;
typedef __attribute__((ext_vector_type(8))) float v8f;

#define POINTS 11
#define PI_F 3.14159265358979323846f
#define NB 2048
#define NT 256

// ---------------- tiny table kernel: softmax rows + WMMA matvec ------------
// ws[0..15]  = eff_theta (11 valid)
// ws[16..31] = eff_vel   (11 valid)
__global__ __launch_bounds__(32) void k_tables(const float* __restrict__ w,
                                               float* __restrict__ ws) {
    __shared__ float sm[POINTS][12];
    const int lane = threadIdx.x;

    if (lane < POINTS) {
        float row[POINTS];
        float mx = -INFINITY;
        for (int j = 0; j < POINTS; ++j) {
            row[j] = w[lane * POINTS + j];
            mx = fmaxf(mx, row[j]);
        }
        float s = 0.f;
        for (int j = 0; j < POINTS; ++j) {
            row[j] = __expf(row[j] - mx);
            s += row[j];
        }
        for (int j = 0; j < POINTS; ++j) sm[lane][j] = row[j] / s;
    }
    __syncthreads();

    // A-matrix 16x4 f32 layout (ISA 7.12.2):
    //   VGPR0: lanes0-15 -> K=k0+0, lanes16-31 -> K=k0+2
    //   VGPR1: lanes0-15 -> K=k0+1, lanes16-31 -> K=k0+3
    const int m = lane & 15;
    const int khalf = (lane >> 4) * 2;      // 0 for lanes 0-15, 2 for 16-31

    const float step = (2.f * PI_F) / (POINTS - 1);
    const float tparam = -PI_F + (float)m * step;   // theta_tab[m]
    const float vparam = 0.3f * (float)m;           // vel_tab[m]
    const bool mok = (m < POINTS);

    v8f ct = {};   // theta accumulator (C=0 initially)
    v8f cv = {};   // vel accumulator
    v2f bone; bone[0] = 1.f; bone[1] = 1.f;  // B = ones(4x16): layout-invariant

    for (int k0 = 0; k0 < 12; k0 += 4) {
        const int ka = k0 + khalf;
        const int kb = ka + 1;
        const float sa = (mok && ka < POINTS) ? sm[m][ka] : 0.f;
        const float sb = (mok && kb < POINTS) ? sm[m][kb] : 0.f;
        v2f at; at[0] = tparam * sa; at[1] = tparam * sb;
        v2f av; av[0] = vparam * sa; av[1] = vparam * sb;
        ct = __builtin_amdgcn_wmma_f32_16x16x4_f32(false, at, false, bone,
                                                   (short)0, ct, false, false);
        cv = __builtin_amdgcn_wmma_f32_16x16x4_f32(false, av, false, bone,
                                                   (short)0, cv, false, false);
    }

    // D layout: VGPR v, lanes0-15: M=v (N=lane); lanes16-31: M=v+8 (N=lane-16).
    // Column N=0 lives in lane 0 (M=0..7) and lane 16 (M=8..15).
    if (lane == 0) {
        for (int v = 0; v < 8; ++v) { ws[v] = ct[v]; ws[16 + v] = cv[v]; }
    }
    if (lane == 16) {
        for (int v = 0; v < 8; ++v) { ws[8 + v] = ct[v]; ws[24 + v] = cv[v]; }
    }
}

// ---------------- fixed-order block reduction helpers ----------------------
__device__ __forceinline__ void block_reduce2(float& s, float& q, float* lds) {
    // wave32 butterfly (fixed order), then cross-wave via LDS (fixed order)
    for (int o = 16; o > 0; o >>= 1) {
        s += __shfl_xor(s, o, 32);
        q += __shfl_xor(q, o, 32);
    }
    const int wid = threadIdx.x >> 5;
    const int lane = threadIdx.x & 31;
    const int nw = blockDim.x >> 5;
    if (lane == 0) { lds[2 * wid] = s; lds[2 * wid + 1] = q; }
    __syncthreads();
    if (threadIdx.x == 0) {
        s = lds[0]; q = lds[1];
        for (int i = 1; i < nw; ++i) { s += lds[2 * i]; q += lds[2 * i + 1]; }
    }
}

// ---------------- pass 1: partial sum / sumsq of data ----------------------
__global__ __launch_bounds__(NT) void k_reduce1(const v4f* __restrict__ d,
                                                float* __restrict__ part,
                                                int n4) {
    __shared__ float lds[16];
    float s = 0.f, q = 0.f;
    const int stride = gridDim.x * blockDim.x;
    for (int i = blockIdx.x * blockDim.x + threadIdx.x; i < n4; i += stride) {
        __builtin_prefetch(&d[i + stride], 0, 0);   // global_prefetch_b8
        v4f v = d[i];
        s += v.x; q = fmaf(v.x, v.x, q);
        s += v.y; q = fmaf(v.y, v.y, q);
        s += v.z; q = fmaf(v.z, v.z, q);
        s += v.w; q = fmaf(v.w, v.w, q);
    }
    block_reduce2(s, q, lds);
    if (threadIdx.x == 0) {
        part[2 * blockIdx.x] = s;
        part[2 * blockIdx.x + 1] = q;
    }
}

// ---------------- finalize: partials -> mean, rstd*scale -------------------
__global__ __launch_bounds__(1024) void k_finalize(const float* __restrict__ part,
                                                   int nb,
                                                   const float* __restrict__ scale,
                                                   float* __restrict__ stats,
                                                   float N) {
    __shared__ float ls[1024];
    __shared__ float lq[1024];
    const int t = threadIdx.x;
    float s = 0.f, q = 0.f;
    for (int i = t; i < nb; i += 1024) { s += part[2 * i]; q += part[2 * i + 1]; }
    ls[t] = s; lq[t] = q;
    __syncthreads();
    for (int o = 512; o > 0; o >>= 1) {
        if (t < o) { ls[t] += ls[t + o]; lq[t] += lq[t + o]; }
        __syncthreads();
    }
    if (t == 0) {
        const float S = ls[0], Q = lq[0];
        const float mean = S / N;
        const float var = (Q - S * S / N) / (N - 1.f);   // ddof=1
        const float rstd = 1.f / sqrtf(var);
        stats[0] = mean;
        stats[1] = rstd * scale[0];
    }
}

// ---------------- elementwise chain ----------------------------------------
__device__ __forceinline__ float lerp_tab(const float* __restrict__ t, float idx) {
    const float p0f = floorf(idx);            // idx >= 0 always here
    const float pos = idx - p0f;
    int p0 = (int)p0f;
    p0 = p0 > 10 ? 10 : p0;
    int p1 = p0 + 1;
    p1 = p1 > 10 ? 10 : p1;
    return (1.f - pos) * t[p0] + pos * t[p1];
}

__device__ __forceinline__ float chain(float x,
                                       const float* __restrict__ efft,
                                       const float* __restrict__ effv) {
    // sigmoid(x)*10, branch-free: v_exp_f32 + v_rcp_f32
    const float e = __expf(-x);
    const float it = 10.f * __builtin_amdgcn_rcpf(1.f + e);
    // |tanh(x)|*10 via odd symmetry: ea = exp(-2|x|) in (0,1] (no overflow)
    const float ea = __expf(-2.f * fabsf(x));
    const float iv = 10.f * (1.f - ea) * __builtin_amdgcn_rcpf(1.f + ea);

    const float th = lerp_tab(efft, it);
    const float ds = lerp_tab(effv, iv);
    float sn, cs;
    __sincosf(th, &sn, &cs);
    return fmaf(x, __expf(ds * sn), ds * cs);         // x*exp(dy)+dx
}

// ---------------- pass 2: map + partial stats of result --------------------
__global__ __launch_bounds__(NT) void k_map(const v4f* __restrict__ d,
                                            v4f* __restrict__ x2,
                                            const float* __restrict__ ws,
                                            float* __restrict__ part,
                                            int n4) {
    __shared__ float efft[16];
    __shared__ float effv[16];
    __shared__ float lds[16];
    if (threadIdx.x < 16) {
        efft[threadIdx.x] = ws[threadIdx.x];
        effv[threadIdx.x] = ws[16 + threadIdx.x];
    }
    __syncthreads();
    const float mean = ws[32];
    const float a = ws[33];                    // rstd1 * iscale

    float s = 0.f, q = 0.f;
    const int stride = gridDim.x * blockDim.x;
    for (int i = blockIdx.x * blockDim.x + threadIdx.x; i < n4; i += stride) {
        v4f v = __builtin_nontemporal_load(&d[i]);    // last use of data
        v4f r;
        {
            float y = chain((v.x - mean) * a, efft, effv);
            r.x = y; s += y; q = fmaf(y, y, q);
        }
        {
            float y = chain((v.y - mean) * a, efft, effv);
            r.y = y; s += y; q = fmaf(y, y, q);
        }
        {
            float y = chain((v.z - mean) * a, efft, effv);
            r.z = y; s += y; q = fmaf(y, y, q);
        }
        {
            float y = chain((v.w - mean) * a, efft, effv);
            r.w = y; s += y; q = fmaf(y, y, q);
        }
        x2[i] = r;                                    // keep in L2: re-read in pass 3
    }
    block_reduce2(s, q, lds);
    if (threadIdx.x == 0) {
        part[2 * blockIdx.x] = s;
        part[2 * blockIdx.x + 1] = q;
    }
}

// ---------------- pass 3: in-place normalize + oscale ----------------------
__global__ __launch_bounds__(NT) void k_out(v4f* __restrict__ io,
                                            const float* __restrict__ ws,
                                            int n4) {
    const float mean = ws[34];
    const float b = ws[35];                    // rstd2 * oscale
    const int stride = gridDim.x * blockDim.x;
    for (int i = blockIdx.x * blockDim.x + threadIdx.x; i < n4; i += stride) {
        v4f v = io[i];
        v.x = (v.x - mean) * b;
        v.y = (v.y - mean) * b;
        v.z = (v.z - mean) * b;
        v.w = (v.w - mean) * b;
        __builtin_nontemporal_store(v, &io[i]);       // never re-read
    }
}

// ---------------- host launcher --------------------------------------------
extern "C" void kernel_launch(void* const* d_in, const int* in_sizes, int n_in,
                              void* d_out, int out_size, void* d_ws, size_t ws_size,
                              hipStream_t stream) {
    const float* data   = (const float*)d_in[0];
    const float* iscale = (const float*)d_in[1];
    const float* oscale = (const float*)d_in[2];
    const float* weight = (const float*)d_in[3];
    float* out = (float*)d_out;
    float* ws  = (float*)d_ws;

    const int n  = in_sizes[0];
    const int n4 = n >> 2;                     // 32*64*128*128 divisible by 4

    // ws layout (floats): [0..31] eff tables, [32..35] stats,
    // [64..64+2*NB) pass-1 partials, [64+2*NB..64+4*NB) pass-2 partials.
    float* part1 = ws + 64;
    float* part2 = ws + 64 + 2 * NB;

    k_tables<<<1, 32, 0, stream>>>(weight, ws);
    k_reduce1<<<NB, NT, 0, stream>>>((const v4f*)data, part1, n4);
    k_finalize<<<1, 1024, 0, stream>>>(part1, NB, iscale, ws + 32, (float)n);
    k_map<<<NB, NT, 0, stream>>>((const v4f*)data, (v4f*)out, ws, part2, n4);
    k_finalize<<<1, 1024, 0, stream>>>(part2, NB, oscale, ws + 34, (float)n);
    k_out<<<NB, NT, 0, stream>>>((v4f*)out, ws, n4);
}